// Translator_71339406787395
// MI455X (gfx1250) — compile-verified
//
#include <hip/hip_runtime.h>
#include <hip/hip_bf16.h>

#define E_N     14541
#define P_N     237
#define V_N     475      // 2*P+1
#define STEPS_N 3
#define B_N     128
#define THR_F   1e-20f
#define TILE_E  256

// ---------------- CDNA5 feature guards (compile-safe fallbacks) -------------
#if __has_builtin(__builtin_amdgcn_global_load_async_to_lds_b32) && \
    __has_builtin(__builtin_amdgcn_s_wait_asynccnt)
#define USE_ASYNC_LDS 1
#else
#define USE_ASYNC_LDS 0
#endif

#if __has_builtin(__builtin_amdgcn_wmma_f32_16x16x4_f32)
#define USE_WMMA_ROWSUM 1
typedef __attribute__((ext_vector_type(2))) float v2f;
typedef __attribute__((ext_vector_type(8))) float v8f;
#endif

#if USE_ASYNC_LDS
// Probe-learned signature: parameters are plain generic 'int *' pointers.
__device__ __forceinline__ void async_b32(const void* g, void* l) {
  __builtin_amdgcn_global_load_async_to_lds_b32((int*)g, (int*)l, 0, 0);
}
#endif

// ---------------- K1: edge filter vals[e] in {0,1} --------------------------
__global__ void k_vals(const int* __restrict__ heads, const int* __restrict__ tails,
                       const int* __restrict__ rels,
                       const int* __restrict__ th, const int* __restrict__ tr,
                       const int* __restrict__ tt,
                       float* __restrict__ vals, int nnz) {
  __shared__ int sh[B_N], sr[B_N], st[B_N];
  if (threadIdx.x < B_N) {
    sh[threadIdx.x] = th[threadIdx.x];
    sr[threadIdx.x] = tr[threadIdx.x];
    st[threadIdx.x] = tt[threadIdx.x];
  }
  __syncthreads();
  int e = blockIdx.x * blockDim.x + threadIdx.x;
  if (e < nnz) {
    int h = heads[e], t = tails[e], r = rels[e];
    float v = 1.0f;
    for (int b = 0; b < B_N; ++b) {
      if (sh[b] == h && st[b] == t && sr[b] == r && sr[b] < P_N) { v = 0.0f; break; }
    }
    vals[e] = v;
  }
}

// ---------------- K2: zero + one-hot init ----------------------------------
__global__ void k_zero(float* __restrict__ p, int n) {
  for (int i = blockIdx.x * blockDim.x + threadIdx.x; i < n; i += gridDim.x * blockDim.x)
    p[i] = 0.0f;
}

__global__ void k_onehot(float* __restrict__ mem_T, const int* __restrict__ th) {
  int b = threadIdx.x;                       // one block of 128
  mem_T[(size_t)th[b] * B_N + b] = 1.0f;
}

// ---------------- K3: transpose attention slice to [V, B] -------------------
__global__ void k_att_T(const float* __restrict__ att, float* __restrict__ att_T, int step) {
  int idx = blockIdx.x * blockDim.x + threadIdx.x;     // over V*B
  if (idx < V_N * B_N) {
    int v = idx / B_N, b = idx - v * B_N;
    att_T[idx] = att[(b * STEPS_N + step) * V_N + v];
  }
}

// ---------------- K4: added = mem * att_last (self-loop); zero rowsum -------
__global__ void k_init_added(const float* __restrict__ mem_T, const float* __restrict__ att_T,
                             float* __restrict__ added_T, float* __restrict__ rowsum) {
  if (blockIdx.x == 0 && threadIdx.x < B_N) rowsum[threadIdx.x] = 0.0f;
  const int n = E_N * B_N;
  for (int i = blockIdx.x * blockDim.x + threadIdx.x; i < n; i += gridDim.x * blockDim.x) {
    int b = i & (B_N - 1);
    added_T[i] = mem_T[i] * att_T[(V_N - 1) * B_N + b];
  }
}

// ---------------- K5: edge gather/scatter (lanes = batches) -----------------
__global__ void k_edges(const int* __restrict__ heads, const int* __restrict__ tails,
                        const int* __restrict__ rels, const float* __restrict__ vals,
                        const float* __restrict__ att_T, const float* __restrict__ mem_T,
                        float* __restrict__ added_T, int nnz) {
  __shared__ int   sh[TILE_E];
  __shared__ int   st[TILE_E];
  __shared__ int   sr[TILE_E];
  __shared__ float sv[TILE_E];
  const int b    = threadIdx.x;                  // 0..127 (4 wave32s)
  const int base = blockIdx.x * TILE_E;
  int n = nnz - base; if (n > TILE_E) n = TILE_E;

#if USE_ASYNC_LDS
  for (int i = threadIdx.x; i < n; i += B_N) {
    int e = base + i;
    async_b32(heads + e, sh + i);
    async_b32(tails + e, st + i);
    async_b32(rels  + e, sr + i);
    async_b32(vals  + e, sv + i);
  }
  __builtin_amdgcn_s_wait_asynccnt(0);
  __syncthreads();
#else
  for (int i = threadIdx.x; i < n; i += B_N) {
    int e = base + i;
    sh[i] = heads[e]; st[i] = tails[e]; sr[i] = rels[e]; sv[i] = vals[e];
  }
  __syncthreads();
#endif

  for (int j = 0; j < n; ++j) {
    if (sv[j] != 0.0f) {                         // uniform branch (LDS scalar)
      const int h = sh[j], t = st[j], r = sr[j];
      float af = att_T[r * B_N + b];
      float ab = att_T[(r + P_N) * B_N + b];
      float mf = mem_T[(size_t)h * B_N + b];
      float mb = mem_T[(size_t)t * B_N + b];
      atomicAdd(added_T + (size_t)t * B_N + b, mf * af);
      atomicAdd(added_T + (size_t)h * B_N + b, mb * ab);
    }
  }
}

// ---------------- K6: rowsum[b] = sum_h added_T[h*B + b] --------------------
// WMMA path: A = ones(16x4) f32, B = 4 entities x 16 batches  =>  exact f32
// column sums accumulate in C across the entity loop.
__global__ void k_rowsum(const float* __restrict__ added_T, float* __restrict__ rowsum) {
  const int EPB = (E_N + gridDim.x - 1) / gridDim.x;
  const int e0 = blockIdx.x * EPB;
  int e1 = e0 + EPB; if (e1 > E_N) e1 = E_N;
  if (e0 >= e1) return;
#if USE_WMMA_ROWSUM
  const int wave  = threadIdx.x >> 5;            // 0..7 (blockDim=256)
  const int lane  = threadIdx.x & 31;
  const int half  = lane >> 4;                   // 0/1
  const int col   = lane & 15;
  const int batch = wave * 16 + col;             // 8 waves x 16 = 128 batches
  v2f a; a[0] = 1.0f; a[1] = 1.0f;
  v8f c = {};
  for (int e = e0; e < e1; e += 4) {             // uniform trip count: EXEC all-1s
    int ea = e + half * 2;
    int eb = ea + 1;
    v2f bv;
    bv[0] = (ea < e1) ? added_T[(size_t)ea * B_N + batch] : 0.0f;
    bv[1] = (eb < e1) ? added_T[(size_t)eb * B_N + batch] : 0.0f;
    c = __builtin_amdgcn_wmma_f32_16x16x4_f32(false, a, false, bv, (short)0, c,
                                              false, false);
  }
  if (half == 0) atomicAdd(rowsum + batch, c[0]); // D[0,n] = colsum, lanes 0..15
#else
  const int b   = threadIdx.x & (B_N - 1);
  const int sub = threadIdx.x >> 7;              // 0/1
  float acc = 0.0f;
  for (int e = e0 + sub; e < e1; e += 2) acc += added_T[(size_t)e * B_N + b];
  atomicAdd(rowsum + b, acc);
#endif
}

// ---------------- K7: normalize into mem_T ---------------------------------
__global__ void k_norm(const float* __restrict__ added_T, const float* __restrict__ rowsum,
                       float* __restrict__ mem_T) {
  const int n = E_N * B_N;
  for (int i = blockIdx.x * blockDim.x + threadIdx.x; i < n; i += gridDim.x * blockDim.x) {
    int b = i & (B_N - 1);
    mem_T[i] = added_T[i] / fmaxf(THR_F, rowsum[b]);
  }
}

// ---------------- K8: write mem_final [B,E] + loss --------------------------
__global__ void k_out(const float* __restrict__ mem_T, float* __restrict__ out) {
  int idx = blockIdx.x * blockDim.x + threadIdx.x;       // over B*E
  if (idx < B_N * E_N) {
    int b = idx / E_N;
    int h = idx - b * E_N;
    out[1 + idx] = mem_T[(size_t)h * B_N + b];
  }
}

__global__ void k_loss(const float* __restrict__ mem_T, const int* __restrict__ tt,
                       float* __restrict__ out) {
  __shared__ float s[B_N];
  int b = threadIdx.x;
  float sc = mem_T[(size_t)tt[b] * B_N + b];
  s[b] = logf(fmaxf(THR_F, sc));
  __syncthreads();
  for (int o = B_N >> 1; o > 0; o >>= 1) {
    if (b < o) s[b] += s[b + o];
    __syncthreads();
  }
  if (b == 0) out[0] = -s[0] / (float)B_N;
}

// ---------------- launcher --------------------------------------------------
extern "C" void kernel_launch(void* const* d_in, const int* in_sizes, int n_in,
                              void* d_out, int out_size, void* d_ws, size_t ws_size,
                              hipStream_t stream) {
  const float* att  = (const float*)d_in[0];
  const int* heads  = (const int*)d_in[1];
  const int* tails  = (const int*)d_in[2];
  const int* rels   = (const int*)d_in[3];
  const int* th     = (const int*)d_in[4];
  const int* tr     = (const int*)d_in[5];
  const int* tt     = (const int*)d_in[6];
  float* out        = (float*)d_out;
  const int nnz     = in_sizes[1];

  float* ws      = (float*)d_ws;
  float* mem_T   = ws;                                    // E*B
  float* added_T = mem_T   + (size_t)E_N * B_N;           // E*B
  float* att_T   = added_T + (size_t)E_N * B_N;           // V*B
  float* vals    = att_T   + (size_t)V_N * B_N;           // NNZ
  float* rowsum  = vals    + (size_t)nnz;                 // B

  k_vals<<<(nnz + 255) / 256, 256, 0, stream>>>(heads, tails, rels, th, tr, tt, vals, nnz);
  k_zero<<<2048, 256, 0, stream>>>(mem_T, E_N * B_N);
  k_onehot<<<1, B_N, 0, stream>>>(mem_T, th);

  for (int s = 0; s < STEPS_N; ++s) {
    k_att_T<<<(V_N * B_N + 255) / 256, 256, 0, stream>>>(att, att_T, s);
    k_init_added<<<2048, 256, 0, stream>>>(mem_T, att_T, added_T, rowsum);
    k_edges<<<(nnz + TILE_E - 1) / TILE_E, B_N, 0, stream>>>(heads, tails, rels, vals,
                                                             att_T, mem_T, added_T, nnz);
    k_rowsum<<<128, 256, 0, stream>>>(added_T, rowsum);
    k_norm<<<2048, 256, 0, stream>>>(added_T, rowsum, mem_T);
  }

  k_out<<<(B_N * E_N + 255) / 256, 256, 0, stream>>>(mem_T, out);
  k_loss<<<1, B_N, 0, stream>>>(mem_T, tt, out);
}